// PyramidROIAlign_69801808495427
// MI455X (gfx1250) — compile-verified
//
#include <hip/hip_runtime.h>
#include <stdint.h>

typedef uint32_t u32;
typedef __attribute__((ext_vector_type(4))) u32   u32x4;
typedef __attribute__((ext_vector_type(8))) u32   u32x8;
typedef __attribute__((ext_vector_type(4))) float f32x4;

#define PH   7
#define PW   7
#define CCH  256
#define TILE_F (PH * PW * CCH)   /* 12544 floats = 50176 bytes, contiguous per box */

__launch_bounds__(256)
__global__ void roi_align_tdm(const float* __restrict__ boxes,
                              const float* __restrict__ p3,
                              const float* __restrict__ p4,
                              const float* __restrict__ p5,
                              float* __restrict__ out,
                              int N)
{
    __shared__ __align__(16) float tile[TILE_F];

    const int b   = blockIdx.y;              // batch index (no integer division)
    const int box = b * N + blockIdx.x;      // flat b*N + n

    // ---- box -> pyramid level (block-uniform -> scalarized by compiler) ----
    const float y1 = boxes[box * 4 + 0];
    const float x1 = boxes[box * 4 + 1];
    const float y2 = boxes[box * 4 + 2];
    const float x2 = boxes[box * 4 + 3];
    const float h = y2 - y1, w = x2 - x1;
    float lvlf = ceilf(5.0f + __log2f(h * w));
    lvlf = fminf(fmaxf(lvlf, 3.0f), 5.0f);
    const int lvl = (int)lvlf;

    const float* feat;
    int S;
    if (lvl == 3)      { feat = p3; S = 128; }
    else if (lvl == 4) { feat = p4; S = 64;  }
    else               { feat = p5; S = 32;  }

    const float Sm1 = (float)(S - 1);
    const float sy = h * Sm1 * (1.0f / 6.0f);
    const float sx = w * Sm1 * (1.0f / 6.0f);
    const float oy = y1 * Sm1;
    const float ox = x1 * Sm1;

    const f32x4* __restrict__ f4 = (const f32x4*)(feat + (size_t)b * S * S * CCH);
    f32x4* t4 = (f32x4*)tile;

    // ---- bilinear gather: 49 pixels x 64 float4 channel-chunks ----
    for (int u = threadIdx.x; u < PH * PW * (CCH / 4); u += 256) {
        const int c4  = u & 63;       // channel chunk (16B) -> coalesced lanes
        const int pix = u >> 6;       // 0..48
        const int iy  = pix / PW;
        const int ix  = pix - iy * PW;

        const float ys = oy + (float)iy * sy;
        const float xs = ox + (float)ix * sx;
        const float yf = floorf(ys), xf = floorf(xs);
        const float wy = ys - yf,    wx = xs - xf;

        const int y0i = (int)fminf(fmaxf(yf,        0.0f), Sm1);
        const int y1i = (int)fminf(fmaxf(yf + 1.0f, 0.0f), Sm1);
        const int x0i = (int)fminf(fmaxf(xf,        0.0f), Sm1);
        const int x1i = (int)fminf(fmaxf(xf + 1.0f, 0.0f), Sm1);
        const bool valid = (ys >= 0.0f) && (ys <= Sm1) && (xs >= 0.0f) && (xs <= Sm1);

        const int r0 = y0i * S, r1 = y1i * S;
        const f32x4 g00 = f4[(size_t)(r0 + x0i) * (CCH / 4) + c4];
        const f32x4 g01 = f4[(size_t)(r0 + x1i) * (CCH / 4) + c4];
        const f32x4 g10 = f4[(size_t)(r1 + x0i) * (CCH / 4) + c4];
        const f32x4 g11 = f4[(size_t)(r1 + x1i) * (CCH / 4) + c4];

        const f32x4 top = g00 + wx * (g01 - g00);
        const f32x4 bot = g10 + wx * (g11 - g10);
        f32x4 o = top + wy * (bot - top);
        const f32x4 zero = {0.0f, 0.0f, 0.0f, 0.0f};
        if (!valid) o = zero;

        t4[u] = o;                    // ds_store_b128 into the staging tile
    }
    __syncthreads();                  // tile fully assembled in LDS

    // ---- one wave issues a single TDM store: LDS tile -> contiguous 50KB in d_out ----
    const int wave = __builtin_amdgcn_readfirstlane((int)(threadIdx.x >> 5));
    if (wave == 0) {
        const u32      lds_off = (u32)(uintptr_t)tile;  // flat LDS addr low 32b == LDS offset
        const uint64_t ga = (uint64_t)(uintptr_t)(out + (size_t)box * TILE_F);

        // D# group 0 (128b): count=1 | lds_addr | global_addr[56:0] | type=2
        u32x4 g0;
        g0[0] = 1u;                                        // count=1, user mode
        g0[1] = lds_off;                                   // lds_addr (bytes)
        g0[2] = (u32)ga;                                   // global_addr[31:0]
        g0[3] = ((u32)(ga >> 32) & 0x01FFFFFFu) | (2u << 30); // global_addr[56:32] | type=2

        // D# group 1 (256b): 2D tensor, data_size=4B, one row of 12544 elements
        u32x8 g1;
        g1[0] = (2u << 16);                                // workgroup_mask=0, data_size=2 (4B)
        g1[1] = ((u32)TILE_F & 0xFFFFu) << 16;             // tensor_dim0[15:0] @ bits 63:48
        g1[2] = ((u32)TILE_F >> 16) | (1u << 16);          // tensor_dim0[31:16] | tensor_dim1=1
        g1[3] = ((u32)TILE_F) << 16;                       // tensor_dim1 hi=0 | tile_dim0 @ 127:112
        g1[4] = 1u;                                        // tile_dim1=1, tile_dim2=0
        g1[5] = (u32)TILE_F;                               // tensor_dim0_stride[31:0]
        g1[6] = 0u;                                        // stride0 hi, stride1 lo
        g1[7] = 0u;                                        // stride1 hi

        // Output is write-once / never re-read: stream it past L2 (keep the
        // L2-resident feature maps hot) with a non-temporal store hint.
        asm volatile("tensor_store_from_lds %0, %1 th:TH_STORE_NT"
                     :: "s"(g0), "s"(g1)
                     : "memory");
        __builtin_amdgcn_s_wait_tensorcnt(0);
    }
}

extern "C" void kernel_launch(void* const* d_in, const int* in_sizes, int n_in,
                              void* d_out, int out_size, void* d_ws, size_t ws_size,
                              hipStream_t stream) {
    // setup_inputs order: boxes, positive_indices, p3, p4, p5, config
    const float* boxes = (const float*)d_in[0];
    const float* p3    = (const float*)d_in[2];
    const float* p4    = (const float*)d_in[3];
    const float* p5    = (const float*)d_in[4];
    float* out = (float*)d_out;

    const int nBoxes = in_sizes[0] / 4;                 // B*N
    int B = in_sizes[2] / (128 * 128 * 256);
    if (B <= 0) B = 1;
    const int N = nBoxes / B;

    dim3 grid((unsigned)N, (unsigned)B, 1);
    roi_align_tdm<<<grid, 256, 0, stream>>>(boxes, p3, p4, p5, out, N);
}